// UniformViewTransformer_87419764343043
// MI455X (gfx1250) — compile-verified
//
#include <hip/hip_runtime.h>
#include <hip/hip_bf16.h>
#include <math.h>

typedef float v2f __attribute__((ext_vector_type(2)));
typedef float v8f __attribute__((ext_vector_type(8)));

#define NVOX   131072      // 8*128*128
#define NCAM   6
#define FH     32
#define FW     88
#define FC     128
#define HW     (FH*FW)     // 2816

// ---------------------------------------------------------------------------
// Kernel 0: P[c][i][k] = sum_j cam2img[c][i][j] * ego2cam[c][j][k], i=0..2
// Stored flat as P[(c*3+i)*4 + k]  (18 rows x 4), the A-matrix rows for WMMA.
// ---------------------------------------------------------------------------
__global__ void projmat_kernel(const float* __restrict__ e2c,
                               const float* __restrict__ c2i,
                               float* __restrict__ P) {
  int t = threadIdx.x;
  if (t < 72) {
    int c = t / 12;
    int i = (t % 12) >> 2;
    int j = t & 3;
    float s = 0.f;
#pragma unroll
    for (int k = 0; k < 4; ++k)
      s += c2i[c * 16 + i * 4 + k] * e2c[c * 16 + k * 4 + j];
    P[t] = s;
  }
}

// ---------------------------------------------------------------------------
// Kernel 1: transpose features (6,128,32,88) -> (6,32,88,128) so that the 128
// channels of one pixel are contiguous (512B) => float4 loads in the fuser.
// ---------------------------------------------------------------------------
__global__ __launch_bounds__(256) void transpose_kernel(const float* __restrict__ in,
                                                        float* __restrict__ outT) {
  int t = blockIdx.x * 256 + threadIdx.x;        // 0 .. 2162687
  int c = t & 127;
  int x = (t >> 7) % FW;
  int r = t / (FC / 128 * 128 * FW / 128 * 128); // placeholder avoided below
  r = t / (FW * FC);                              // cam*32 + y
  outT[t] = in[(size_t)((r >> 5) * 128 + c) * HW + (size_t)(r & 31) * FW + x];
}

// ---------------------------------------------------------------------------
// Kernel 2: projection GEMM via V_WMMA_F32_16X16X4_F32.
//   D[m][n] = sum_k A[m][k] * B[k][n]
//   A rows  = proj rows (block0: cams 0-3 -> rows 0..11; block1: cams 4,5 -> 0..5)
//   B cols  = 16 voxels' homogeneous coords (x,y,z,1)
// A f32 16x4 layout: lanes 0-15 = M, VGPR0/1 = K0/K1; lanes 16-31 = K2/K3.
// B f32 4x16 layout (mirrored): VGPR0 = K0|K2, VGPR1 = K1|K3 across lane halves.
// D layout: VGPR r, lanes 0-15 -> (M=r, N=lane); lanes 16-31 -> (M=r+8).
// Raw D is dumped to ws: projR[b16*512 + blk*256 + r*32 + lane].
// ---------------------------------------------------------------------------
__global__ __launch_bounds__(256) void proj_wmma_kernel(const float* __restrict__ P,
                                                        float* __restrict__ projR) {
  int lane = threadIdx.x & 31;
  int waveBase = blockIdx.x * 256 + (threadIdx.x & ~31);  // 32 voxels per wave
  int half = lane >> 4;
  int mrow = lane & 15;

  v2f a0, a1;
  {
    int k0 = half * 2, k1 = half * 2 + 1;
    a0.x = (mrow < 12) ? P[mrow * 4 + k0] : 0.f;
    a0.y = (mrow < 12) ? P[mrow * 4 + k1] : 0.f;
    a1.x = (mrow < 6) ? P[(12 + mrow) * 4 + k0] : 0.f;
    a1.y = (mrow < 6) ? P[(12 + mrow) * 4 + k1] : 0.f;
  }

#pragma unroll
  for (int g = 0; g < 2; ++g) {
    int vox = waveBase + g * 16 + (lane & 15);
    float wx = -50.8f + 0.8f * (float)(vox & 127);
    float wy = -50.8f + 0.8f * (float)((vox >> 7) & 127);
    float wz = -4.5f + (float)(vox >> 14);
    v2f b;
    b.x = half ? wz : wx;    // K=2 : K=0
    b.y = half ? 1.0f : wy;  // K=3 : K=1
    v8f cz = {};
    v8f d0 = __builtin_amdgcn_wmma_f32_16x16x4_f32(false, a0, false, b,
                                                   (short)0, cz, false, false);
    v8f d1 = __builtin_amdgcn_wmma_f32_16x16x4_f32(false, a1, false, b,
                                                   (short)0, cz, false, false);
    size_t o = (size_t)((waveBase >> 4) + g) * 512 + lane;
#pragma unroll
    for (int r = 0; r < 8; ++r) {
      projR[o + r * 32] = d0[r];
      projR[o + 256 + r * 32] = d1[r];
    }
  }
}

// ---------------------------------------------------------------------------
// Kernel 3: fuse. One lane = one voxel. Reads (u*d, v*d, d) per cam from the
// WMMA dump, builds 4 premultiplied corner weights (depth + bounds folded in,
// base pixel clamped so loads are always in-bounds), then 32 channel-quad
// iterations of 6 cams x 4 aligned float4 loads and 4 coalesced stores.
// ---------------------------------------------------------------------------
__global__ __launch_bounds__(256) void fuse_kernel(const float* __restrict__ featT,
                                                   const float* __restrict__ projR,
                                                   float* __restrict__ out) {
  int vox = blockIdx.x * 256 + threadIdx.x;
  int n = vox & 15;
  const float* pr = projR + (size_t)(vox >> 4) * 512;

  float w00[NCAM], w01[NCAM], w10[NCAM], w11[NCAM];
  int base[NCAM];
  float cnt = 0.f;

#pragma unroll
  for (int c = 0; c < NCAM; ++c) {
    const int blk = (c < 4) ? 0 : 256;
    const int m0 = (c < 4) ? 3 * c : 3 * (c - 4);
    // slot(m) = (m&7)*32 + (m>>3)*16 + n
    float ud = pr[blk + ((m0 & 7) << 5) + ((m0 >> 3) << 4) + n];
    float vd = pr[blk + (((m0 + 1) & 7) << 5) + (((m0 + 1) >> 3) << 4) + n];
    float d  = pr[blk + (((m0 + 2) & 7) << 5) + (((m0 + 2) >> 3) << 4) + n];

    float dm = (d > 0.1f) ? 1.f : 0.f;
    cnt += dm;
    float inv = 1.0f / (d + 1e-5f);
    float u = ud * inv, v = vd * inv;
    float x0 = floorf(u), y0 = floorf(v);
    float wx1 = u - x0, wy1 = v - y0;
    float wx0 = 1.f - wx1, wy0 = 1.f - wy1;
    bool vx0 = (x0 >= 0.f) && (x0 <= (float)(FW - 1));
    bool vx1 = (x0 >= -1.f) && (x0 <= (float)(FW - 2));
    bool vy0 = (y0 >= 0.f) && (y0 <= (float)(FH - 1));
    bool vy1 = (y0 >= -1.f) && (y0 <= (float)(FH - 2));
    w00[c] = (vx0 && vy0) ? wx0 * wy0 * dm : 0.f;
    w01[c] = (vx1 && vy0) ? wx1 * wy0 * dm : 0.f;
    w10[c] = (vx0 && vy1) ? wx0 * wy1 * dm : 0.f;
    w11[c] = (vx1 && vy1) ? wx1 * wy1 * dm : 0.f;
    int bx = (int)fminf(fmaxf(x0, 0.f), (float)(FW - 2));
    int by = (int)fminf(fmaxf(y0, 0.f), (float)(FH - 2));
    base[c] = ((c * FH + by) * FW + bx) * FC;   // float index into featT
  }

  float invc = 1.0f / (cnt + 1e-6f);

#pragma unroll 2
  for (int q = 0; q < 32; ++q) {
    float a0 = 0.f, a1 = 0.f, a2 = 0.f, a3 = 0.f;
#pragma unroll
    for (int c = 0; c < NCAM; ++c) {
      const float4* p = (const float4*)(featT + base[c] + q * 4);
      float4 s00 = p[0];
      float4 s01 = p[FC / 4];            // +1 pixel in x  (+128 floats)
      float4 s10 = p[FW * FC / 4];       // +1 pixel in y
      float4 s11 = p[(FW + 1) * FC / 4]; // +1 in x and y
      a0 += w00[c] * s00.x + w01[c] * s01.x + w10[c] * s10.x + w11[c] * s11.x;
      a1 += w00[c] * s00.y + w01[c] * s01.y + w10[c] * s10.y + w11[c] * s11.y;
      a2 += w00[c] * s00.z + w01[c] * s01.z + w10[c] * s10.z + w11[c] * s11.z;
      a3 += w00[c] * s00.w + w01[c] * s01.w + w10[c] * s10.w + w11[c] * s11.w;
    }
    size_t ob = (size_t)(q * 4) * NVOX + vox;
    out[ob] = a0 * invc;
    out[ob + NVOX] = a1 * invc;
    out[ob + 2 * (size_t)NVOX] = a2 * invc;
    out[ob + 3 * (size_t)NVOX] = a3 * invc;
  }
}

// ---------------------------------------------------------------------------
// Fallback: single monolithic kernel (no workspace), per-lane FMA projection,
// original feature layout.
// ---------------------------------------------------------------------------
__global__ __launch_bounds__(256) void fuse_fallback(const float* __restrict__ feats,
                                                     const float* __restrict__ e2c,
                                                     const float* __restrict__ c2i,
                                                     float* __restrict__ out) {
  int vox = blockIdx.x * 256 + threadIdx.x;
  float wx = -50.8f + 0.8f * (float)(vox & 127);
  float wy = -50.8f + 0.8f * (float)((vox >> 7) & 127);
  float wz = -4.5f + (float)(vox >> 14);

  float w00[NCAM], w01[NCAM], w10[NCAM], w11[NCAM];
  int base[NCAM];
  float cnt = 0.f;

#pragma unroll
  for (int c = 0; c < NCAM; ++c) {
    float q[4];
#pragma unroll
    for (int k = 0; k < 4; ++k)
      q[k] = e2c[c * 16 + k * 4 + 0] * wx + e2c[c * 16 + k * 4 + 1] * wy +
             e2c[c * 16 + k * 4 + 2] * wz + e2c[c * 16 + k * 4 + 3];
    float ud = 0.f, vd = 0.f, d = 0.f;
#pragma unroll
    for (int k = 0; k < 4; ++k) {
      ud += c2i[c * 16 + 0 * 4 + k] * q[k];
      vd += c2i[c * 16 + 1 * 4 + k] * q[k];
      d  += c2i[c * 16 + 2 * 4 + k] * q[k];
    }
    float dm = (d > 0.1f) ? 1.f : 0.f;
    cnt += dm;
    float inv = 1.0f / (d + 1e-5f);
    float u = ud * inv, v = vd * inv;
    float x0 = floorf(u), y0 = floorf(v);
    float wx1 = u - x0, wy1 = v - y0;
    float wx0 = 1.f - wx1, wy0 = 1.f - wy1;
    bool vx0 = (x0 >= 0.f) && (x0 <= (float)(FW - 1));
    bool vx1 = (x0 >= -1.f) && (x0 <= (float)(FW - 2));
    bool vy0 = (y0 >= 0.f) && (y0 <= (float)(FH - 1));
    bool vy1 = (y0 >= -1.f) && (y0 <= (float)(FH - 2));
    w00[c] = (vx0 && vy0) ? wx0 * wy0 * dm : 0.f;
    w01[c] = (vx1 && vy0) ? wx1 * wy0 * dm : 0.f;
    w10[c] = (vx0 && vy1) ? wx0 * wy1 * dm : 0.f;
    w11[c] = (vx1 && vy1) ? wx1 * wy1 * dm : 0.f;
    int bx = (int)fminf(fmaxf(x0, 0.f), (float)(FW - 2));
    int by = (int)fminf(fmaxf(y0, 0.f), (float)(FH - 2));
    base[c] = c * FC * HW + by * FW + bx;
  }
  float invc = 1.0f / (cnt + 1e-6f);
  for (int ch = 0; ch < FC; ++ch) {
    float acc = 0.f;
#pragma unroll
    for (int c = 0; c < NCAM; ++c) {
      const float* p = feats + base[c] + ch * HW;
      acc += w00[c] * p[0] + w01[c] * p[1] + w10[c] * p[FW] + w11[c] * p[FW + 1];
    }
    out[(size_t)ch * NVOX + vox] = acc * invc;
  }
}

extern "C" void kernel_launch(void* const* d_in, const int* in_sizes, int n_in,
                              void* d_out, int out_size, void* d_ws, size_t ws_size,
                              hipStream_t stream) {
  const float* img = (const float*)d_in[0];  // (6,128,32,88)
  const float* e2c = (const float*)d_in[1];  // (6,4,4)
  const float* c2i = (const float*)d_in[2];  // (6,4,4)
  float* out = (float*)d_out;                // (1,128,8,128,128)

  const size_t projFloats = (size_t)(NVOX / 16) * 512;   // 4,194,304
  const size_t featFloats = (size_t)NCAM * FH * FW * FC; // 2,162,688
  const size_t needBytes = (128 + projFloats + featFloats) * sizeof(float);

  if (ws_size >= needBytes) {
    float* P = (float*)d_ws;
    float* projR = P + 128;
    float* featT = projR + projFloats;

    projmat_kernel<<<1, 128, 0, stream>>>(e2c, c2i, P);
    transpose_kernel<<<(int)(featFloats / 256), 256, 0, stream>>>(img, featT);
    proj_wmma_kernel<<<NVOX / 256, 256, 0, stream>>>(P, projR);
    fuse_kernel<<<NVOX / 256, 256, 0, stream>>>(featT, projR, out);
  } else {
    fuse_fallback<<<NVOX / 256, 256, 0, stream>>>(img, e2c, c2i, out);
  }
}